// RegionLoss_82995948028354
// MI455X (gfx1250) — compile-verified
//
#include <hip/hip_runtime.h>
#include <hip/hip_bf16.h>
#include <math.h>

// ---------------------------------------------------------------------------
// YOLOv2 region loss for gfx1250 (MI455X).
//   output: (16, 5*85, 64, 64) fp32, target: (16, 250) fp32 -> scalar loss.
// Design:
//   k_prep   : 1 block. Per-(b,t) GT preprocessing (valid cumprod, anchor
//              argmax, scatter key, tx/ty/tw/th/tconf/tcls, nGT).
//   k_main   : 1280 blocks x 256. Reads ONLY channels 0..4 per anchor
//              (sigmoids / exp / no-div IoU threshold / GT match); reads the
//              80 class channels only at masked cells (<=800 threads total).
//              Unmasked rows contribute exactly log(80) to the class loss.
//              Block-reduces to one fp32 partial per block.
//   k_reduce : 1 wave. Deterministic fixed-order sum of the 1280 partials on
//              the matrix pipe via v_wmma_f32_16x16x4_f32 (B = ones ->
//              exact fp32 row sums accumulated into C), then /nGT.
//              1280 == 20*64, so all loads are unconditionally in-bounds:
//              no exec-mask guards around the WMMA feed loads.
// ---------------------------------------------------------------------------

typedef __attribute__((ext_vector_type(2))) float v2f;
typedef __attribute__((ext_vector_type(8))) float v8f;

#if defined(__has_builtin)
#if __has_builtin(__builtin_amdgcn_wmma_f32_16x16x4_f32)
#define HAVE_WMMA_F32_16x16x4 1
#endif
#endif

#define NB 16
#define NA 5
#define NC 80
#define NH 64
#define NW 64
#define HW (NH * NW)       // 4096
#define MAXGT 50
#define NCELL_BLOCKS (NB * NA * (HW / 256))   // 1280 (multiple of 64)
#define LOG80 4.3820266346738812f

// Workspace layout (float/int element offsets into d_ws).
enum : int {
  OF_X1  = 0,       // gt x-left            [800]
  OF_X2  = 800,     // gt x-right           [800]
  OF_Y1  = 1600,    // gt y-top             [800]
  OF_Y2  = 2400,    // gt y-bottom          [800]
  OF_W   = 3200,    // gt w                 [800]
  OF_H   = 4000,    // gt h                 [800]
  OF_AR  = 4800,    // gt area              [800]
  OF_KEY = 5600,    // int: a*4096+jj*64+ii [800]
  OF_TX  = 6400,    // tx                   [800]
  OF_TY  = 7200,    // ty                   [800]
  OF_TW  = 8000,    // tw                   [800]
  OF_TH  = 8800,    // th                   [800]
  OF_TC  = 9600,    // tconf (anchor iou)   [800]
  OF_CL  = 10400,   // int: class id        [800]
  OF_NV  = 11200,   // int: nvalid per b    [16]
  OF_NGT = 11232,   // float: nGT           [1]
  OF_PART= 11264,   // block partials       [1280]
};

__constant__ float c_aw[NA] = {1.3221f, 3.19275f, 5.05587f, 9.47112f, 11.2364f};
__constant__ float c_ah[NA] = {1.73145f, 4.00944f, 8.09892f, 4.84053f, 10.0071f};

// ------------------------------ kernel 1 -----------------------------------
__global__ void k_prep(const float* __restrict__ tgt, float* __restrict__ ws) {
  __shared__ int s_nv[NB];
  int b = threadIdx.x;
  if (b < NB) {
    const float* t = tgt + b * (MAXGT * 5);
    int* wsi = (int*)ws;
    int nv = MAXGT;
    for (int i = 0; i < MAXGT; i++) {
      float cls = t[i * 5 + 0];
      float x   = t[i * 5 + 1];
      float y   = t[i * 5 + 2];
      float w   = t[i * 5 + 3];
      float h   = t[i * 5 + 4];
      bool valid = (i < nv) && (x != 0.0f);   // cumprod semantics
      if (!valid && i < nv) nv = i;
      int gi = b * MAXGT + i;
      if (valid) {
        float gx = x * (float)NW, gy = y * (float)NH;
        float gw = w * (float)NW, gh = h * (float)NH;
        // anchor argmax (centered IoU of shapes)
        float best = -1.0f; int bn = 0;
        for (int a = 0; a < NA; a++) {
          float inter = fminf(gw, c_aw[a]) * fminf(gh, c_ah[a]);
          float uni   = gw * gh + c_aw[a] * c_ah[a] - inter;
          float iou   = inter / uni;
          if (iou > best) { best = iou; bn = a; }
        }
        int ii = (int)floorf(gx), jj = (int)floorf(gy);
        ws[OF_X1 + gi] = gx - gw * 0.5f;
        ws[OF_X2 + gi] = gx + gw * 0.5f;
        ws[OF_Y1 + gi] = gy - gh * 0.5f;
        ws[OF_Y2 + gi] = gy + gh * 0.5f;
        ws[OF_W  + gi] = gw;
        ws[OF_H  + gi] = gh;
        ws[OF_AR + gi] = gw * gh;
        wsi[OF_KEY + gi] = bn * 4096 + jj * 64 + ii;
        ws[OF_TX + gi] = gx - (float)ii;
        ws[OF_TY + gi] = gy - (float)jj;
        ws[OF_TW + gi] = logf(gw / c_aw[bn]);
        ws[OF_TH + gi] = logf(gh / c_ah[bn]);
        ws[OF_TC + gi] = best;
        wsi[OF_CL + gi] = (int)cls;
      } else {
        ws[OF_X1 + gi] = 0.0f; ws[OF_X2 + gi] = 0.0f;
        ws[OF_Y1 + gi] = 0.0f; ws[OF_Y2 + gi] = 0.0f;
        ws[OF_W  + gi] = 0.0f; ws[OF_H  + gi] = 0.0f;
        ws[OF_AR + gi] = 0.0f;
        wsi[OF_KEY + gi] = -1;
        ws[OF_TX + gi] = 0.0f; ws[OF_TY + gi] = 0.0f;
        ws[OF_TW + gi] = 0.0f; ws[OF_TH + gi] = 0.0f;
        ws[OF_TC + gi] = 0.0f;
        wsi[OF_CL + gi] = 0;
      }
    }
    ((int*)ws)[OF_NV + b] = nv;
    s_nv[b] = nv;
  }
  __syncthreads();
  if (threadIdx.x == 0) {
    int tot = 0;
    for (int i = 0; i < NB; i++) tot += s_nv[i];
    ws[OF_NGT] = (float)tot;
  }
}

// ------------------------------ kernel 2 -----------------------------------
__global__ void __launch_bounds__(256) k_main(const float* __restrict__ O,
                                              float* __restrict__ ws) {
  __shared__ float sx1[MAXGT], sx2[MAXGT], sy1[MAXGT], sy2[MAXGT];
  __shared__ float sgw[MAXGT], sgh[MAXGT], sar[MAXGT];
  __shared__ int   skey[MAXGT];
  __shared__ float stx[MAXGT], sty[MAXGT], stw[MAXGT], sth[MAXGT], stc[MAXGT];
  __shared__ int   scl[MAXGT];
  __shared__ float sred[256];

  int bx    = blockIdx.x;
  int plane = bx >> 4;              // b*NA + a
  int tile  = bx & 15;
  int b     = plane / NA;
  int a     = plane - b * NA;
  int tid   = threadIdx.x;

  if (tid < MAXGT) {
    int gi = b * MAXGT + tid;
    const int* wsi = (const int*)ws;
    sx1[tid] = ws[OF_X1 + gi];  sx2[tid] = ws[OF_X2 + gi];
    sy1[tid] = ws[OF_Y1 + gi];  sy2[tid] = ws[OF_Y2 + gi];
    sgw[tid] = ws[OF_W  + gi];  sgh[tid] = ws[OF_H  + gi];
    sar[tid] = ws[OF_AR + gi];
    skey[tid] = wsi[OF_KEY + gi];
    stx[tid] = ws[OF_TX + gi];  sty[tid] = ws[OF_TY + gi];
    stw[tid] = ws[OF_TW + gi];  sth[tid] = ws[OF_TH + gi];
    stc[tid] = ws[OF_TC + gi];
    scl[tid] = wsi[OF_CL + gi];
  }
  __syncthreads();

  int nv = ((const int*)ws)[OF_NV + b];
  int hw = tile * 256 + tid;
  int h  = hw >> 6;
  int w  = hw & 63;

  const float* base = O + (size_t)plane * 85 * HW + hw;
  float o0 = base[0];
  float o1 = base[1 * HW];
  float o2 = base[2 * HW];
  float o3 = base[3 * HW];
  float o4 = base[4 * HW];

  float sigx = 1.0f / (1.0f + expf(-o0));
  float sigy = 1.0f / (1.0f + expf(-o1));
  float conf = 1.0f / (1.0f + expf(-o4));
  float aw = c_aw[a], ah = c_ah[a];
  float pw = expf(o2) * aw;
  float ph = expf(o3) * ah;
  float px = sigx + (float)w;
  float py = sigy + (float)h;
  float px1 = px - pw * 0.5f, px2 = px + pw * 0.5f;
  float py1 = py - ph * 0.5f, py2 = py + ph * 0.5f;
  float parea = pw * ph;

  // best_iou_cell > 0.6  <=>  exists t: inter > 0.6*union   (no division)
  bool sil = false;
  int  tHit = -1;
  int  myKey = a * 4096 + hw;
  for (int t = 0; t < nv; t++) {
    float mx = fminf(px1, sx1[t]);
    float Mx = fmaxf(px2, sx2[t]);
    float my = fminf(py1, sy1[t]);
    float My = fmaxf(py2, sy2[t]);
    float cw = pw + sgw[t] - (Mx - mx);
    float ch = ph + sgh[t] - (My - my);
    float inter = (cw > 0.0f && ch > 0.0f) ? cw * ch : 0.0f;
    float uni = parea + sar[t] - inter;
    sil = sil || (inter > 0.6f * uni);
    if (skey[t] == myKey) tHit = t;   // last valid GT wins (scatter .set)
  }

  float partial;
  if (tHit >= 0) {
    // object cell: coord + 25*(conf-tconf)^2 + (lse - logit[tcls])
    const float* cbase = base + 5 * HW;
    __builtin_prefetch(cbase, 0, 0);                 // global_prefetch_b8
    float dx = sigx - stx[tHit];
    float dy = sigy - sty[tHit];
    float dw = o2   - stw[tHit];
    float dh = o3   - sth[tHit];
    float dc = conf - stc[tHit];
    float m = -3.402823466e38f;
    for (int c = 0; c < NC; c++) m = fmaxf(m, cbase[c * HW]);
    float se = 0.0f;
    for (int c = 0; c < NC; c++) se += expf(cbase[c * HW] - m);
    float lse = m + logf(se);
    partial = dx * dx + dy * dy + dw * dw + dh * dh
            + 25.0f * dc * dc
            + (lse - cbase[scl[tHit] * HW]);
  } else {
    // no-object cell: scale in {0,1}; masked logits are all-zero -> log(80)
    float scale = sil ? 0.0f : 1.0f;
    partial = scale * conf * conf + LOG80;
  }

  sred[tid] = partial;
  __syncthreads();
  for (int s = 128; s > 0; s >>= 1) {
    if (tid < s) sred[tid] += sred[tid + s];
    __syncthreads();
  }
  if (tid == 0) ws[OF_PART + bx] = sred[0];
}

// ------------------------------ kernel 3 -----------------------------------
// One wave. Sum NCELL_BLOCKS partials with v_wmma_f32_16x16x4_f32 (B = ones):
// per issue A holds 64 values (A[m][k], m = lane%16, k = 2*(lane/16)+vgpr),
// D[m][n] = sum_k A[m][k] + C[m][n] -> exact fp32 row sums accumulated in C.
// Total = sum over m of D[m][0] = c-sum(lane 0) + c-sum(lane 16).
// NCELL_BLOCKS is a multiple of 64 -> loads are unconditional (no exec-mask
// guards around the WMMA feed loads).
__global__ void k_reduce(const float* __restrict__ ws_c, float* __restrict__ ws,
                         float* __restrict__ out) {
  const float* partials = ws_c + OF_PART;
  int lane = threadIdx.x;
  int col  = lane & 15;
  int half = lane >> 4;
  __shared__ float sh[32];

#if defined(HAVE_WMMA_F32_16x16x4)
  static_assert((NCELL_BLOCKS & 63) == 0, "reduce loop assumes 64-multiples");
  v8f acc = {};
  v2f bone; bone.x = 1.0f; bone.y = 1.0f;
  const float* p = partials + half * 32 + col;
  for (int basei = 0; basei < NCELL_BLOCKS; basei += 64) {
    v2f av;
    av.x = p[basei];           // (k = 2*half + 0, m = col)
    av.y = p[basei + 16];      // (k = 2*half + 1, m = col)
    acc = __builtin_amdgcn_wmma_f32_16x16x4_f32(
        /*neg_a=*/false, av, /*neg_b=*/false, bone,
        /*c_mod=*/(short)0, acc, /*reuse_a=*/false, /*reuse_b=*/false);
  }
  float s = acc[0] + acc[1] + acc[2] + acc[3] +
            acc[4] + acc[5] + acc[6] + acc[7];
  sh[lane] = s;
  __syncthreads();
  if (lane == 0) {
    float total = sh[0] + sh[16];          // rows 0..7 + rows 8..15 at n=0
    out[0] = total / ws[OF_NGT];
  }
#else
  // Fallback: serial deterministic sum (should not be needed on gfx1250).
  if (lane == 0) {
    float total = 0.0f;
    for (int i = 0; i < NCELL_BLOCKS; i++) total += partials[i];
    out[0] = total / ws[OF_NGT];
  }
  (void)col; (void)half; (void)sh;
#endif
}

// ------------------------------ launcher -----------------------------------
extern "C" void kernel_launch(void* const* d_in, const int* in_sizes, int n_in,
                              void* d_out, int out_size, void* d_ws, size_t ws_size,
                              hipStream_t stream) {
  const float* output = (const float*)d_in[0];   // (16, 425, 64, 64) fp32
  const float* target = (const float*)d_in[1];   // (16, 250)         fp32
  float* ws  = (float*)d_ws;                     // needs ~50 KB
  float* out = (float*)d_out;                    // 1 fp32 scalar

  k_prep<<<1, 64, 0, stream>>>(target, ws);
  k_main<<<NCELL_BLOCKS, 256, 0, stream>>>(output, ws);
  k_reduce<<<1, 32, 0, stream>>>(ws, ws, out);
}